// DecoderRNN_69020124447165
// MI455X (gfx1250) — compile-verified
//
#include <hip/hip_runtime.h>
#include <math.h>

#define Vdim 8192
#define Hdim 1024
#define Bdim 64
#define Tdim 128
#define G4H  4096   // 4*H

typedef float v2f __attribute__((ext_vector_type(2)));
typedef float v8f __attribute__((ext_vector_type(8)));

__device__ __forceinline__ float sigmoidf_(float x) {
    return 1.0f / (1.0f + __expf(-x));
}

// D = A (64 x 1024, row-major) * W^T (W is N x 1024, row-major) + bias1 [+ bias2]
// Output element (m, n) goes to D[dOff + m*rowStride + n*colStride].
// One wave handles 4 m-tiles x 2 n-tiles; 4 waves per block -> 8 n-tiles/block.
__global__ __launch_bounds__(128) void gemm_wmma_f32(
    const float* __restrict__ A,
    const float* __restrict__ W,
    const float* __restrict__ bias1,
    const float* __restrict__ bias2,   // may be null
    float* __restrict__ D,
    long rowStride, long colStride, long dOff)
{
    const int lane = threadIdx.x & 31;
    const int wave = threadIdx.x >> 5;
    const int lr   = lane & 15;          // row/col within a 16-tile
    const int kh   = (lane >> 4) << 1;   // K sub-offset: 0 (lanes 0-15) or 2 (lanes 16-31)
    const int n0   = (blockIdx.x * 8 + wave * 2) << 4;  // first of two n-tiles

    // A-operand pointers for the 4 m-tiles (M = mt*16 + lr)
    const float* a0 = A + (0 * 16 + lr) * Hdim + kh;
    const float* a1 = A + (1 * 16 + lr) * Hdim + kh;
    const float* a2 = A + (2 * 16 + lr) * Hdim + kh;
    const float* a3 = A + (3 * 16 + lr) * Hdim + kh;
    // B-operand pointers for the 2 n-tiles (W stored N x K row-major)
    const float* b0 = W + (long)(n0      + lr) * Hdim + kh;
    const float* b1 = W + (long)(n0 + 16 + lr) * Hdim + kh;

    v8f acc00 = {}; v8f acc01 = {};
    v8f acc10 = {}; v8f acc11 = {};
    v8f acc20 = {}; v8f acc21 = {};
    v8f acc30 = {}; v8f acc31 = {};

    for (int k = 0; k < Hdim; k += 4) {
        v2f vb0 = *(const v2f*)(b0 + k);
        v2f vb1 = *(const v2f*)(b1 + k);
        v2f va0 = *(const v2f*)(a0 + k);
        v2f va1 = *(const v2f*)(a1 + k);
        v2f va2 = *(const v2f*)(a2 + k);
        v2f va3 = *(const v2f*)(a3 + k);
        acc00 = __builtin_amdgcn_wmma_f32_16x16x4_f32(false, va0, false, vb0, (short)0, acc00, false, false);
        acc01 = __builtin_amdgcn_wmma_f32_16x16x4_f32(false, va0, false, vb1, (short)0, acc01, false, false);
        acc10 = __builtin_amdgcn_wmma_f32_16x16x4_f32(false, va1, false, vb0, (short)0, acc10, false, false);
        acc11 = __builtin_amdgcn_wmma_f32_16x16x4_f32(false, va1, false, vb1, (short)0, acc11, false, false);
        acc20 = __builtin_amdgcn_wmma_f32_16x16x4_f32(false, va2, false, vb0, (short)0, acc20, false, false);
        acc21 = __builtin_amdgcn_wmma_f32_16x16x4_f32(false, va2, false, vb1, (short)0, acc21, false, false);
        acc30 = __builtin_amdgcn_wmma_f32_16x16x4_f32(false, va3, false, vb0, (short)0, acc30, false, false);
        acc31 = __builtin_amdgcn_wmma_f32_16x16x4_f32(false, va3, false, vb1, (short)0, acc31, false, false);
    }

    // Epilogue: D layout per 16x16 f32 tile: VGPR r holds M = r + 8*(lane>=16), N = lane&15
    const int mhi = (lane >> 4) << 3;   // 0 or 8
    v8f accs[4][2] = { {acc00, acc01}, {acc10, acc11}, {acc20, acc21}, {acc30, acc31} };
#pragma unroll
    for (int n = 0; n < 2; ++n) {
        const int col = n0 + n * 16 + lr;
        float bv = bias1[col];
        if (bias2) bv += bias2[col];
#pragma unroll
        for (int m = 0; m < 4; ++m) {
            const int rowBase = m * 16 + mhi;
#pragma unroll
            for (int r = 0; r < 8; ++r) {
                D[dOff + (long)(rowBase + r) * rowStride + (long)col * colStride] =
                    accs[m][n][r] + bv;
            }
        }
    }
}

// Fused: gather-accumulate previous argmax column of W_ih into gates, then LSTM cell.
// One thread per (b, hidden-index). At t==0, gates already hold hh and c starts at 0.
__global__ __launch_bounds__(256) void lstm_step(
    const float* __restrict__ W_ih,   // (4H, V) row-major
    float* __restrict__ gates,        // (B, 4H) running accumulator (starts at hh)
    float* __restrict__ c,            // (B, H)
    float* __restrict__ h,            // (B, H)
    const int* __restrict__ idx,      // (B) argmax of previous step
    int t)
{
    const int i  = blockIdx.x * blockDim.x + threadIdx.x;   // 0 .. B*H-1
    const int b  = i >> 10;           // / Hdim
    const int hx = i & (Hdim - 1);

    float* g = gates + (long)b * G4H;
    float ig = g[hx];
    float fg = g[Hdim + hx];
    float gg = g[2 * Hdim + hx];
    float og = g[3 * Hdim + hx];

    if (t > 0) {
        const int id = idx[b];
        ig += W_ih[(long)(hx)            * Vdim + id];
        fg += W_ih[(long)(Hdim + hx)     * Vdim + id];
        gg += W_ih[(long)(2 * Hdim + hx) * Vdim + id];
        og += W_ih[(long)(3 * Hdim + hx) * Vdim + id];
        g[hx]            = ig;
        g[Hdim + hx]     = fg;
        g[2 * Hdim + hx] = gg;
        g[3 * Hdim + hx] = og;
    }

    const float cprev = (t == 0) ? 0.0f : c[i];
    const float cn = sigmoidf_(fg) * cprev + sigmoidf_(ig) * tanhf(gg);
    c[i] = cn;
    h[i] = sigmoidf_(og) * tanhf(cn);
}

// Argmax over V for one batch row per block (first-max/lowest-index tie-break, as jnp.argmax).
__global__ __launch_bounds__(256) void argmax_step(
    const float* __restrict__ out,    // d_out, (B, V, T)
    int* __restrict__ idx,
    int t)
{
    const int b   = blockIdx.x;
    const int tid = threadIdx.x;
    __shared__ float sval[256];
    __shared__ int   sidx[256];

    float best = -INFINITY;
    int   bi   = Vdim;
    for (int v = tid; v < Vdim; v += 256) {
        const float x = out[((long)b * Vdim + v) * Tdim + t];
        if (x > best || (x == best && v < bi)) { best = x; bi = v; }
    }
    sval[tid] = best;
    sidx[tid] = bi;
    __syncthreads();
    for (int s = 128; s > 0; s >>= 1) {
        if (tid < s) {
            const float ov = sval[tid + s];
            const int   oi = sidx[tid + s];
            if (ov > sval[tid] || (ov == sval[tid] && oi < sidx[tid])) {
                sval[tid] = ov; sidx[tid] = oi;
            }
        }
        __syncthreads();
    }
    if (tid == 0) idx[b] = sidx[0];
}

extern "C" void kernel_launch(void* const* d_in, const int* in_sizes, int n_in,
                              void* d_out, int out_size, void* d_ws, size_t ws_size,
                              hipStream_t stream) {
    const float* h0   = (const float*)d_in[0];
    const float* W_ih = (const float*)d_in[1];
    const float* W_hh = (const float*)d_in[2];
    const float* b_ih = (const float*)d_in[3];
    const float* b_hh = (const float*)d_in[4];
    const float* W_fc = (const float*)d_in[5];
    const float* b_fc = (const float*)d_in[6];
    float* out = (float*)d_out;

    float* ws    = (float*)d_ws;
    float* gates = ws;                          // B*4H
    float* c     = gates + (long)Bdim * G4H;    // B*H
    float* h     = c     + (long)Bdim * Hdim;   // B*H
    int*   idx   = (int*)(h + (long)Bdim * Hdim);

    // gates(init) = hh = h0 @ W_hh^T + b_ih + b_hh   (M=64, N=4096, K=1024)
    gemm_wmma_f32<<<G4H / 128, 128, 0, stream>>>(
        h0, W_hh, b_ih, b_hh, gates, (long)G4H, 1L, 0L);

    for (int t = 0; t < Tdim; ++t) {
        // gather W_ih column of previous argmax into gates, run LSTM cell
        lstm_step<<<(Bdim * Hdim) / 256, 256, 0, stream>>>(W_ih, gates, c, h, idx, t);
        // out[:, :, t] = h @ W_fc^T + b_fc   (M=64, N=8192, K=1024), fp32 WMMA
        gemm_wmma_f32<<<Vdim / 128, 128, 0, stream>>>(
            h, W_fc, b_fc, nullptr, out, (long)Vdim * Tdim, (long)Tdim, (long)t);
        // argmax over V feeds next step
        argmax_step<<<Bdim, 256, 0, stream>>>(out, idx, t);
    }
}